// MaskAlignmentLoss_37271726195151
// MI455X (gfx1250) — compile-verified
//
#include <hip/hip_runtime.h>
#include <hip/hip_bf16.h>
#include <math.h>

typedef float v2f __attribute__((ext_vector_type(2)));
typedef float v8f __attribute__((ext_vector_type(8)));

#define B_  4
#define N_  6890
#define H_  64
#define W_  64
#define HW_ (H_ * W_)
#define NTILES (HW_ / 16)            // 256 pixel tiles of 16

#define WAVES_PER_WG 8
#define THREADS_PER_WG (WAVES_PER_WG * 32)
#define VERTS_PER_WG (WAVES_PER_WG * 16)                       // 128
#define WGS_PER_BATCH ((N_ + VERTS_PER_WG - 1) / VERTS_PER_WG) // 54

// Dynamic LDS layout: [B-operand table: NTILES*32 float2 = 64KB][pixel mins: HW_ uints = 16KB]
#define LDS_BTAB_BYTES (NTILES * 32 * sizeof(float) * 2)
#define LDS_TOTAL_BYTES (LDS_BTAB_BYTES + HW_ * sizeof(unsigned int))

#define FLT_MAX_BITS 0x7F7FFFFFu

// ---------------------------------------------------------------------------
// Kernel 0: init global per-pixel min workspace and the output scalar.
// ---------------------------------------------------------------------------
__global__ void mal_init(unsigned int* __restrict__ pixmin, float* __restrict__ out) {
    int idx = blockIdx.x * blockDim.x + threadIdx.x;
    if (idx < B_ * HW_) pixmin[idx] = FLT_MAX_BITS;
    if (idx == 0) out[0] = 0.0f;
}

// ---------------------------------------------------------------------------
// Kernel 1: WMMA chamfer tiles.
//   A row (vert m)  = (-2vx, -2vy, |v|^2, 1)
//   B col (pixel p) = ( px ,  py , 1, |p|^2 + mask_penalty)
//   D = A x B  ->  16x16 tile of exact f32 squared distances.
// A layout (16x4 f32, 2 VGPRs): lanes 0-15 -> K=0,1 ; lanes 16-31 -> K=2,3.
// B layout (4x16 f32, 2 VGPRs): lane n holds col n; lanes 0-15 -> K=0,1,
//                               lanes 16-31 -> K=2,3.  Staged in LDS in exactly
//                               this layout so the hot loop is one ds_load_b64.
// D layout (8 VGPRs): vgpr r -> M=r (lanes 0-15) / M=r+8 (lanes 16-31), N=lane&15.
// ---------------------------------------------------------------------------
__global__ __launch_bounds__(THREADS_PER_WG)
void mal_wmma(const float* __restrict__ vert2d, const int* __restrict__ mask,
              unsigned int* __restrict__ pixmin, float* __restrict__ out) {
    extern __shared__ unsigned char smem[];
    v2f* __restrict__ bTab            = (v2f*)smem;
    unsigned int* __restrict__ pixLds = (unsigned int*)(smem + LDS_BTAB_BYTES);

    const int lane  = threadIdx.x & 31;
    const int wave  = threadIdx.x >> 5;
    const int b     = blockIdx.x / WGS_PER_BATCH;
    const int vbase = (blockIdx.x % WGS_PER_BATCH) * VERTS_PER_WG + wave * 16;

    const int m  = lane & 15;
    const int hi = lane >> 4;

    const int* __restrict__ mrow = mask + b * HW_;

    // ---- Phase A: build B-operand table + init pixel-min LDS (once per WG) ----
    for (int e = threadIdx.x; e < NTILES * 32; e += THREADS_PER_WG) {
        const int t = e >> 5;
        const int l = e & 31;
        const int n = l & 15;
        const int h = l >> 4;
        const int p = (t << 4) | n;
        const float px = (float)(p & (W_ - 1)) * (1.0f / W_);
        const float py = (float)(p >> 6)       * (1.0f / H_);
        const float pen = (mrow[p] > 0) ? 0.0f : 1.0e18f;  // exclude invalid pixels from vert mins
        v2f v;
        v.x = h ? 1.0f : px;
        v.y = h ? (px * px + py * py + pen) : py;
        bTab[e] = v;
    }
    for (int i = threadIdx.x; i < HW_; i += THREADS_PER_WG)
        pixLds[i] = FLT_MAX_BITS;
    __syncthreads();

    // ---- Build A operand (this wave's 16 verts), branchless ----
    const int  vi     = vbase + m;
    const bool vvalid = (vi < N_);
    const int  vc     = vvalid ? vi : (N_ - 1);
    const float vx = vert2d[(b * N_ + vc) * 2 + 0] * (1.0f / W_);
    const float vy = vert2d[(b * N_ + vc) * 2 + 1] * (1.0f / H_);
    v2f a;
    a.x = hi ? (vvalid ? (vx * vx + vy * vy) : 1.0e30f) : (-2.0f * vx);
    a.y = hi ? 1.0f : (-2.0f * vy);

    v8f vmin;
    #pragma unroll
    for (int r = 0; r < 8; ++r) vmin[r] = 3.0e38f;

    v8f c;
    #pragma unroll
    for (int r = 0; r < 8; ++r) c[r] = 0.0f;

    // ---- Phase B: hot loop — ds_load + wmma + mins, no branches ----
    #pragma unroll 4
    for (int t = 0; t < NTILES; ++t) {
        const v2f bb = bTab[t * 32 + lane];                    // ds_load_b64, conflict-free

        v8f d = __builtin_amdgcn_wmma_f32_16x16x4_f32(
            /*neg_a=*/false, a, /*neg_b=*/false, bb,
            /*c_mod=*/(short)0, c, /*reuse_a=*/false, /*reuse_b=*/false);

        // pixel-side: min over this wave's 16 verts.  Lanes n and n+16 cover
        // M=0-7 and M=8-15; both ds_min the same pixel slot (no shuffle needed).
        float pm = fminf(d[0], d[1]);
        pm = fminf(pm, fminf(d[2], d[3]));
        pm = fminf(pm, fminf(d[4], d[5]));
        pm = fminf(pm, fminf(d[6], d[7]));
        pm = fmaxf(pm, 0.0f);                                  // clamp FP noise; keeps uint-min valid
        atomicMin(&pixLds[(t << 4) | m], __float_as_uint(pm)); // ds_min_u32

        // vert-side: running min over pixels
        #pragma unroll
        for (int r = 0; r < 8; ++r) vmin[r] = fminf(vmin[r], d[r]);
    }

    // ---- Phase C: vert-side reduction across the 16 N-lanes of each half ----
    #pragma unroll
    for (int r = 0; r < 8; ++r) {
        #pragma unroll
        for (int s = 1; s < 16; s <<= 1)
            vmin[r] = fminf(vmin[r], __shfl_xor(vmin[r], s, 32));
    }
    if (m == 0) {   // lanes 0 and 16 hold the 8 vert mins for M = r + 8*hi
        float s = 0.0f;
        #pragma unroll
        for (int r = 0; r < 8; ++r) {
            int vm = vbase + r + 8 * hi;
            if (vm < N_) s += sqrtf(fmaxf(vmin[r], 0.0f));
        }
        atomicAdd(out, s);
    }

    // ---- Phase D: merge this WG's pixel mins into the global table ----
    __syncthreads();
    unsigned int* __restrict__ gmin = pixmin + b * HW_;
    for (int i = threadIdx.x; i < HW_; i += THREADS_PER_WG)
        atomicMin(&gmin[i], pixLds[i]);
}

// ---------------------------------------------------------------------------
// Kernel 2: sum sqrt(per-pixel min d^2) over valid pixels.
// ---------------------------------------------------------------------------
__global__ __launch_bounds__(256)
void mal_pixsum(const int* __restrict__ mask, const unsigned int* __restrict__ pixmin,
                float* __restrict__ out) {
    int idx = blockIdx.x * blockDim.x + threadIdx.x;
    float s = 0.0f;
    if (idx < B_ * HW_ && mask[idx] > 0)
        s = sqrtf(__uint_as_float(pixmin[idx]));

    #pragma unroll
    for (int off = 16; off > 0; off >>= 1) s += __shfl_xor(s, off, 32);

    __shared__ float red[256 / 32];
    if ((threadIdx.x & 31) == 0) red[threadIdx.x >> 5] = s;
    __syncthreads();
    if (threadIdx.x == 0) {
        float t = 0.0f;
        #pragma unroll
        for (int w = 0; w < 256 / 32; ++w) t += red[w];
        atomicAdd(out, t);
    }
}

// ---------------------------------------------------------------------------
extern "C" void kernel_launch(void* const* d_in, const int* in_sizes, int n_in,
                              void* d_out, int out_size, void* d_ws, size_t ws_size,
                              hipStream_t stream) {
    (void)in_sizes; (void)n_in; (void)out_size; (void)ws_size;

    const float* vert2d  = (const float*)d_in[0];   // [B, N, 2] f32
    const int*   mask    = (const int*)d_in[1];     // [B, H, W] i32
    float*       out     = (float*)d_out;           // scalar loss
    unsigned int* pixmin = (unsigned int*)d_ws;     // B*HW uints (64 KB)

    const int npix = B_ * HW_;

    mal_init<<<(npix + 255) / 256, 256, 0, stream>>>(pixmin, out);

    mal_wmma<<<B_ * WGS_PER_BATCH, THREADS_PER_WG, LDS_TOTAL_BYTES, stream>>>(
        vert2d, mask, pixmin, out);

    mal_pixsum<<<(npix + 255) / 256, 256, 0, stream>>>(mask, pixmin, out);
}